// DiGCN_InceptionBlock_50491635532092
// MI455X (gfx1250) — compile-verified
//
#include <hip/hip_runtime.h>

typedef __attribute__((ext_vector_type(2))) float v2f;
typedef __attribute__((ext_vector_type(8))) float v8f;

#define NNODES 50000
#define DDIM   128
#define NEDGES 600000

// ---------------------------------------------------------------------------
// GEMM via V_WMMA_F32_16X16X4_F32 (fp32 in, fp32 accumulate).
// One wave -> 16 rows x 128 cols of the output.
// BT=true : out = X @ W^T  (W is [Dout, Din], row-major)  -> B loads contiguous
// BT=false: out = X @ W    (W is [Din, Dout], row-major)
// bias (len D) added on store if non-null.
// ---------------------------------------------------------------------------
template<bool BT>
__global__ __launch_bounds__(256)
void gemm_wmma(const float* __restrict__ X,
               const float* __restrict__ W,
               const float* __restrict__ bias,
               float* __restrict__ out,
               int nrows) {
  const int wave = threadIdx.x >> 5;      // 0..7
  const int lane = threadIdx.x & 31;
  const int half = lane >> 4;             // 0 or 1 (selects K pair)
  const int l16  = lane & 15;
  const int m0   = (blockIdx.x * 8 + wave) * 16;
  if (m0 >= nrows) return;                // wave-uniform: EXEC stays all-ones

  // Clamp row for A loads in the tail block (duplicates are harmless; stores
  // below are predicated per-row).
  int mrow = m0 + l16;
  if (mrow >= nrows) mrow = nrows - 1;
  const float* xrow = X + (size_t)mrow * DDIM;

  v8f acc[8];
#pragma unroll
  for (int t = 0; t < 8; ++t) acc[t] = (v8f)0.f;

  for (int kk = 0; kk < DDIM; kk += 4) {
    const int ka = kk + 2 * half;         // this half-wave's K pair
    // A 16x4 f32 layout: lane -> M, v0/v1 -> K=ka, ka+1
    v2f a = *(const v2f*)(xrow + ka);
#pragma unroll
    for (int t = 0; t < 8; ++t) {
      const int n = t * 16 + l16;         // lane -> N within this column tile
      v2f b;
      if (BT) {
        // B[k][n] = W[n][k] -> contiguous pair in W row n
        b = *(const v2f*)(W + (size_t)n * DDIM + ka);
      } else {
        b.x = W[(size_t)ka       * DDIM + n];
        b.y = W[(size_t)(ka + 1) * DDIM + n];
      }
      // D = A(16x4) x B(4x16) + C
      acc[t] = __builtin_amdgcn_wmma_f32_16x16x4_f32(
          /*neg_a=*/false, a, /*neg_b=*/false, b,
          /*c_mod=*/(short)0, acc[t], /*reuse_a=*/false, /*reuse_b=*/false);
    }
  }

  // C/D layout: VGPR v, lanes 0-15 -> M=v, lanes 16-31 -> M=v+8; lane%16 -> N.
#pragma unroll
  for (int t = 0; t < 8; ++t) {
    const int n  = t * 16 + l16;
    const float bv = bias ? bias[n] : 0.f;
#pragma unroll
    for (int v = 0; v < 8; ++v) {
      const int m = m0 + v + 8 * half;
      if (m < nrows) out[(size_t)m * DDIM + n] = acc[t][v] + bv;
    }
  }
}

// ---------------------------------------------------------------------------
// Initialize an output region to its (broadcast) bias vector.
// ---------------------------------------------------------------------------
__global__ __launch_bounds__(256)
void init_bias(float* __restrict__ out, const float* __restrict__ b, int total) {
  int i = blockIdx.x * blockDim.x + threadIdx.x;
  if (i < total) out[i] = b[i & (DDIM - 1)];
}

// ---------------------------------------------------------------------------
// Edge scatter: out[dst] += w_e * h[src]. One wave per edge; each lane handles
// 4 features (float4 gather from L2-resident h, 4 f32 atomic adds to out).
// ---------------------------------------------------------------------------
__global__ __launch_bounds__(256)
void scatter_edges(const long long* __restrict__ ei,   // [2, E] int64
                   const float* __restrict__ ew,       // [E]
                   const float* __restrict__ h,        // [N, D]
                   float* __restrict__ out,            // [N, D]
                   int nedges) {
  const int e = (int)((blockIdx.x * blockDim.x + threadIdx.x) >> 5);
  if (e >= nedges) return;
  const int lane = threadIdx.x & 31;

  const long long src = ei[e];
  const long long dst = ei[(size_t)nedges + e];
  const float w = ew[e];

  const float4 hv = *(const float4*)(h + (size_t)src * DDIM + lane * 4);
  float* op = out + (size_t)dst * DDIM + lane * 4;
  atomicAdd(op + 0, w * hv.x);
  atomicAdd(op + 1, w * hv.y);
  atomicAdd(op + 2, w * hv.z);
  atomicAdd(op + 3, w * hv.w);
}

// ---------------------------------------------------------------------------
extern "C" void kernel_launch(void* const* d_in, const int* in_sizes, int n_in,
                              void* d_out, int out_size, void* d_ws, size_t ws_size,
                              hipStream_t stream) {
  const float*     x      = (const float*)d_in[0];
  const long long* ei1    = (const long long*)d_in[1];
  const float*     ew1    = (const float*)d_in[2];
  const long long* ei2    = (const long long*)d_in[3];
  const float*     ew2    = (const float*)d_in[4];
  const float*     ln_w   = (const float*)d_in[5];
  const float*     ln_b   = (const float*)d_in[6];
  const float*     c1_w   = (const float*)d_in[7];
  const float*     c1_b   = (const float*)d_in[8];
  const float*     c2_w   = (const float*)d_in[9];
  const float*     c2_b   = (const float*)d_in[10];

  float* out = (float*)d_out;
  float* x0  = out;                       // [N, D]
  float* x1  = out + (size_t)NNODES * DDIM;
  float* x2  = out + (size_t)2 * NNODES * DDIM;

  float* h1 = (float*)d_ws;               // [N, D] scratch
  float* h2 = h1 + (size_t)NNODES * DDIM;

  const int gemm_blocks = (NNODES + 127) / 128;   // 8 waves x 16 rows per block

  // x0 = x @ ln_w^T + ln_b   (fused bias)
  gemm_wmma<true><<<gemm_blocks, 256, 0, stream>>>(x, ln_w, ln_b, x0, NNODES);
  // h1 = x @ conv1_w ; h2 = x @ conv2_w
  gemm_wmma<false><<<gemm_blocks, 256, 0, stream>>>(x, c1_w, nullptr, h1, NNODES);
  gemm_wmma<false><<<gemm_blocks, 256, 0, stream>>>(x, c2_w, nullptr, h2, NNODES);

  // x1 = conv1_b (broadcast); x2 = conv2_b  — also clears the poisoned output.
  const int tot = NNODES * DDIM;
  init_bias<<<(tot + 255) / 256, 256, 0, stream>>>(x1, c1_b, tot);
  init_bias<<<(tot + 255) / 256, 256, 0, stream>>>(x2, c2_b, tot);

  // x1 += segment_sum(ew1 * h1[src] -> dst) ; same for graph 2.
  const int sc_blocks = (NEDGES * 32 + 255) / 256;  // one wave per edge
  scatter_edges<<<sc_blocks, 256, 0, stream>>>(ei1, ew1, h1, x1, NEDGES);
  scatter_edges<<<sc_blocks, 256, 0, stream>>>(ei2, ew2, h2, x2, NEDGES);
}